// Decoder_26903675142570
// MI455X (gfx1250) — compile-verified
//
#include <hip/hip_runtime.h>
#include <hip/hip_bf16.h>

// Problem constants (from setup_inputs)
#define VV   16000   // vocab
#define EE   256     // embed dim
#define HH   512     // hidden dim
#define BB   8       // batch
#define KB   5       // beams
#define LL   12      // lenseq
#define ROWS 40      // KB*BB
#define RPAD 48      // rows padded to 3x16 WMMA tiles

typedef float v2f __attribute__((ext_vector_type(2)));
typedef float v8f __attribute__((ext_vector_type(8)));

// D = A(16x4 f32) * B(4x16 f32) + C(16x16 f32), full fp32 WMMA
__device__ __forceinline__ v8f wmma4(v2f a, v2f b, v8f c) {
    return __builtin_amdgcn_wmma_f32_16x16x4_f32(false, a, false, b, (short)0, c,
                                                 false, false);
}

// ---------------------------------------------------------------------------
// init: hA rows = broadcast hidden over beams, pad rows zero; tok = SOS
__global__ void init_kernel(const float* __restrict__ hidden,
                            float* __restrict__ hA, int* __restrict__ tok) {
    int row = blockIdx.x;  // 0..47
    for (int u = threadIdx.x; u < HH; u += blockDim.x)
        hA[row * HH + u] = (row < ROWS) ? hidden[(row % BB) * HH + u] : 0.0f;
    if (row < ROWS && threadIdx.x == 0) tok[row] = 1;  // SOS_ID
}

// ---------------------------------------------------------------------------
// per-step prep: done flags + embedding gather
__global__ void prep_kernel(const float* __restrict__ emb, const int* __restrict__ tok,
                            float* __restrict__ embb, int* __restrict__ done_i) {
    int row = blockIdx.x;  // 0..39
    int t = tok[row];
    if (threadIdx.x == 0) done_i[row] = (t == 0 || t == 2) ? 1 : 0;  // PAD|EOS
    const float* er = emb + (long)t * EE;
    for (int c = threadIdx.x; c < EE; c += blockDim.x) embb[row * EE + c] = er[c];
}

// ---------------------------------------------------------------------------
// GRU pre-activation GEMMs via f32 WMMA.
// cols [0,1024): rz = emb@w_ih[c].T + h@w_hh[c].T   (one fused accumulator)
// cols [1024,1536): i_n and h_n kept separate (needed for n = tanh(i_n + r*h_n))
__global__ __launch_bounds__(32) void gru_gemm_wmma(
    const float* __restrict__ embb, const float* __restrict__ hA,
    const float* __restrict__ w_ih, const float* __restrict__ w_hh,
    float* __restrict__ rzbuf, float* __restrict__ inbuf, float* __restrict__ hnbuf) {
    const int lane = threadIdx.x;
    const int n    = lane & 15;   // B col / A row within tile
    const int half = lane >> 4;
    const int gcol = blockIdx.x * 16;   // 0..1535
    const int rt   = blockIdx.y;        // 0..2 row tile
    const float* Ae = embb + (rt * 16 + n) * EE;  // A rows, K=E
    const float* Ah = hA   + (rt * 16 + n) * HH;  // A rows, K=H

    if (gcol < 2 * HH) {  // r and z gates: accumulate both GEMMs together
        v8f acc = {};
        const float* W0 = w_ih + (long)(gcol + n) * EE;
        for (int k0 = 0; k0 < EE; k0 += 4) {
            v2f a  = *(const v2f*)(Ae + k0 + 2 * half);
            v2f bf = *(const v2f*)(W0 + k0 + 2 * half);
            acc = wmma4(a, bf, acc);
        }
        const float* W1 = w_hh + (long)(gcol + n) * HH;
        for (int k0 = 0; k0 < HH; k0 += 4) {
            v2f a  = *(const v2f*)(Ah + k0 + 2 * half);
            v2f bf = *(const v2f*)(W1 + k0 + 2 * half);
            acc = wmma4(a, bf, acc);
        }
#pragma unroll
        for (int v = 0; v < 8; ++v)
            rzbuf[(rt * 16 + v + 8 * half) * (2 * HH) + gcol + n] = acc[v];
    } else {  // n gate: keep i_n and h_n separate
        const int u = gcol - 2 * HH;  // 0..511
        v8f acc1 = {}, acc2 = {};
        const float* W0 = w_ih + (long)(gcol + n) * EE;
        for (int k0 = 0; k0 < EE; k0 += 4) {
            v2f a  = *(const v2f*)(Ae + k0 + 2 * half);
            v2f bf = *(const v2f*)(W0 + k0 + 2 * half);
            acc1 = wmma4(a, bf, acc1);
        }
        const float* W1 = w_hh + (long)(gcol + n) * HH;
        for (int k0 = 0; k0 < HH; k0 += 4) {
            v2f a  = *(const v2f*)(Ah + k0 + 2 * half);
            v2f bf = *(const v2f*)(W1 + k0 + 2 * half);
            acc2 = wmma4(a, bf, acc2);
        }
#pragma unroll
        for (int v = 0; v < 8; ++v) {
            inbuf[(rt * 16 + v + 8 * half) * HH + u + n] = acc1[v];
            hnbuf[(rt * 16 + v + 8 * half) * HH + u + n] = acc2[v];
        }
    }
}

// ---------------------------------------------------------------------------
// Gate nonlinearity: torch GRU order (r,z,n). hhist gets raw h_new; hB gets
// done-masked next-h.
__global__ void gates_kernel(const float* __restrict__ rzbuf,
                             const float* __restrict__ inbuf,
                             const float* __restrict__ hnbuf,
                             const float* __restrict__ hA,
                             const float* __restrict__ b_ih,
                             const float* __restrict__ b_hh,
                             float* __restrict__ hhist_i, float* __restrict__ hB,
                             const int* __restrict__ done_i) {
    int row = blockIdx.x;  // 0..39
    int done = done_i[row];
    for (int u = threadIdx.x; u < HH; u += blockDim.x) {
        float rr = rzbuf[row * (2 * HH) + u]        + b_ih[u]      + b_hh[u];
        float zz = rzbuf[row * (2 * HH) + HH + u]   + b_ih[HH + u] + b_hh[HH + u];
        float in_ = inbuf[row * HH + u] + b_ih[2 * HH + u];
        float hn  = hnbuf[row * HH + u] + b_hh[2 * HH + u];
        float r = 1.0f / (1.0f + expf(-rr));
        float z = 1.0f / (1.0f + expf(-zz));
        float nn = tanhf(in_ + r * hn);
        float hprev = hA[row * HH + u];
        float hnew = (1.0f - z) * nn + z * hprev;
        hhist_i[row * HH + u] = hnew;
        hB[row * HH + u] = done ? hprev : hnew;
    }
}

// ---------------------------------------------------------------------------
// logits = h_new @ w_out.T + b_out, f32 WMMA. One wave per 16-col tile does
// all 3 row tiles, reusing the B fragment in registers.
__global__ __launch_bounds__(32) void logits_gemm_wmma(
    const float* __restrict__ hh, const float* __restrict__ w_out,
    const float* __restrict__ b_out, float* __restrict__ pred) {
    const int lane = threadIdx.x;
    const int n = lane & 15, half = lane >> 4;
    const int col0 = blockIdx.x * 16;
    const float* Wr = w_out + (long)(col0 + n) * HH;
    const float* A0 = hh + (0 * 16 + n) * HH;
    const float* A1 = hh + (1 * 16 + n) * HH;
    const float* A2 = hh + (2 * 16 + n) * HH;
    v8f c0 = {}, c1 = {}, c2 = {};
    for (int k0 = 0; k0 < HH; k0 += 4) {
        v2f bf = *(const v2f*)(Wr + k0 + 2 * half);
        v2f a0 = *(const v2f*)(A0 + k0 + 2 * half);
        v2f a1 = *(const v2f*)(A1 + k0 + 2 * half);
        v2f a2 = *(const v2f*)(A2 + k0 + 2 * half);
        c0 = wmma4(a0, bf, c0);
        c1 = wmma4(a1, bf, c1);
        c2 = wmma4(a2, bf, c2);
    }
    float bias = b_out[col0 + n];
#pragma unroll
    for (int v = 0; v < 8; ++v) {
        pred[(long)( 0 + v + 8 * half) * VV + col0 + n] = c0[v] + bias;
        pred[(long)(16 + v + 8 * half) * VV + col0 + n] = c1[v] + bias;
        pred[(long)(32 + v + 8 * half) * VV + col0 + n] = c2[v] + bias;
    }
}

// ---------------------------------------------------------------------------
// In-place softmax per row; done rows become one_hot(PAD_ID=0).
__global__ void softmax_pred(float* __restrict__ pred, const int* __restrict__ done_i) {
    int row = blockIdx.x;  // 0..39
    __shared__ float red[256];
    float* p = pred + (long)row * VV;
    if (done_i[row]) {
        for (int v = threadIdx.x; v < VV; v += blockDim.x) p[v] = (v == 0) ? 1.0f : 0.0f;
        return;
    }
    float lm = -INFINITY;
    for (int v = threadIdx.x; v < VV; v += blockDim.x) lm = fmaxf(lm, p[v]);
    red[threadIdx.x] = lm; __syncthreads();
    for (int s = 128; s > 0; s >>= 1) {
        if (threadIdx.x < s) red[threadIdx.x] = fmaxf(red[threadIdx.x], red[threadIdx.x + s]);
        __syncthreads();
    }
    float M = red[0]; __syncthreads();
    float ls = 0.0f;
    for (int v = threadIdx.x; v < VV; v += blockDim.x) ls += expf(p[v] - M);
    red[threadIdx.x] = ls; __syncthreads();
    for (int s = 128; s > 0; s >>= 1) {
        if (threadIdx.x < s) red[threadIdx.x] += red[threadIdx.x + s];
        __syncthreads();
    }
    float inv = 1.0f / red[0];
    for (int v = threadIdx.x; v < VV; v += blockDim.x) p[v] = expf(p[v] - M) * inv;
}

// ---------------------------------------------------------------------------
// Exact top-5 per batch over k*V flat scores, jax.lax.top_k tie-break
// (descending value, ascending flat index). Writes backpointers & next tokens,
// then gathers next-step h.
__global__ void topk_kernel(const float* __restrict__ pred, float* __restrict__ hA,
                            const float* __restrict__ hB, int* __restrict__ bp_i,
                            int* __restrict__ tok) {
    const int b = blockIdx.x;
    const int NT = 256;
    __shared__ float sv[NT * 5];
    __shared__ int   si[NT * 5];
    __shared__ int   rbeam[5];
    float tv[5]; int ti[5];
#pragma unroll
    for (int q = 0; q < 5; ++q) { tv[q] = -INFINITY; ti[q] = 0x7FFFFFFF; }
    for (int f = threadIdx.x; f < KB * VV; f += NT) {
        int j = f / VV, v = f - j * VV;
        float val = pred[(long)(j * BB + b) * VV + v];
        if ((val > tv[4]) || (val == tv[4] && f < ti[4])) {
            int q = 4;
            while (q > 0 && ((val > tv[q - 1]) || (val == tv[q - 1] && f < ti[q - 1]))) {
                tv[q] = tv[q - 1]; ti[q] = ti[q - 1]; --q;
            }
            tv[q] = val; ti[q] = f;
        }
    }
#pragma unroll
    for (int q = 0; q < 5; ++q) { sv[threadIdx.x * 5 + q] = tv[q]; si[threadIdx.x * 5 + q] = ti[q]; }
    __syncthreads();
    if (threadIdx.x == 0) {
        for (int q = 0; q < 5; ++q) {
            float bv = -INFINITY; int bf = 0x7FFFFFFF; int bpos = 0;
            for (int t = 0; t < NT * 5; ++t) {
                float v2 = sv[t]; int f2 = si[t];
                if ((v2 > bv) || (v2 == bv && f2 < bf)) { bv = v2; bf = f2; bpos = t; }
            }
            sv[bpos] = -INFINITY; si[bpos] = 0x7FFFFFFF;
            int j = bf / VV, t2 = bf - j * VV;
            bp_i[q * BB + b] = j;
            tok[q * BB + b]  = t2;
            rbeam[q] = j;
        }
    }
    __syncthreads();
    for (int idx = threadIdx.x; idx < KB * HH; idx += NT) {
        int q = idx / HH, u = idx - q * HH;
        hA[(q * BB + b) * HH + u] = hB[(rbeam[q] * BB + b) * HH + u];
    }
}

// ---------------------------------------------------------------------------
// Final: argmax over last pre-reorder pred (first-max tie break == jnp.argmax),
// then walk backpointers to record the chosen ancestor slot per step.
__global__ void final_select(const float* __restrict__ pred, const int* __restrict__ bp,
                             int* __restrict__ chosen) {
    const int b = blockIdx.x;
    __shared__ float rv[256];
    __shared__ int   rf[256];
    float bv = -INFINITY; int bfi = 0x7FFFFFFF;
    for (int f = threadIdx.x; f < KB * VV; f += 256) {
        int j = f / VV, v = f - j * VV;
        float val = pred[(long)(j * BB + b) * VV + v];
        if ((val > bv) || (val == bv && f < bfi)) { bv = val; bfi = f; }
    }
    rv[threadIdx.x] = bv; rf[threadIdx.x] = bfi; __syncthreads();
    for (int s = 128; s > 0; s >>= 1) {
        if (threadIdx.x < s) {
            if ((rv[threadIdx.x + s] > rv[threadIdx.x]) ||
                (rv[threadIdx.x + s] == rv[threadIdx.x] && rf[threadIdx.x + s] < rf[threadIdx.x])) {
                rv[threadIdx.x] = rv[threadIdx.x + s];
                rf[threadIdx.x] = rf[threadIdx.x + s];
            }
        }
        __syncthreads();
    }
    if (threadIdx.x == 0) {
        int s = rf[0] / VV;
        for (int i = LL - 1; i >= 0; --i) {
            int o = bp[i * ROWS + s * BB + b];
            chosen[i * BB + b] = o;
            s = o;
        }
    }
}

// ---------------------------------------------------------------------------
// Reconstruct the winning path's predictions: done -> one_hot(0);
// else softmax(hhist @ w_out.T + b_out). Writes d_out (lenseq,B,V).
__global__ void reconstruct(const float* __restrict__ hhist,
                            const int* __restrict__ chosen,
                            const int* __restrict__ donehist,
                            const float* __restrict__ w_out,
                            const float* __restrict__ b_out,
                            float* __restrict__ out) {
    const int i = blockIdx.x / BB, b = blockIdx.x % BB;
    const int o = chosen[i * BB + b];
    const int row = o * BB + b;
    float* orow = out + (long)(i * BB + b) * VV;
    if (donehist[i * ROWS + row]) {
        for (int v = threadIdx.x; v < VV; v += blockDim.x) orow[v] = (v == 0) ? 1.0f : 0.0f;
        return;
    }
    __shared__ float hrow[HH];
    __shared__ float red[256];
    const float* hp = hhist + (long)(i * RPAD + row) * HH;
    for (int u = threadIdx.x; u < HH; u += blockDim.x) hrow[u] = hp[u];
    __syncthreads();
    float lm = -INFINITY;
    for (int v = threadIdx.x; v < VV; v += blockDim.x) {
        const float* wr = w_out + (long)v * HH;
        float acc = b_out[v];
        for (int e = 0; e < HH; ++e) acc += hrow[e] * wr[e];
        orow[v] = acc;
        lm = fmaxf(lm, acc);
    }
    red[threadIdx.x] = lm; __syncthreads();
    for (int s = 128; s > 0; s >>= 1) {
        if (threadIdx.x < s) red[threadIdx.x] = fmaxf(red[threadIdx.x], red[threadIdx.x + s]);
        __syncthreads();
    }
    float M = red[0]; __syncthreads();
    float ls = 0.0f;
    for (int v = threadIdx.x; v < VV; v += blockDim.x) ls += expf(orow[v] - M);
    red[threadIdx.x] = ls; __syncthreads();
    for (int s = 128; s > 0; s >>= 1) {
        if (threadIdx.x < s) red[threadIdx.x] += red[threadIdx.x + s];
        __syncthreads();
    }
    float inv = 1.0f / red[0];
    for (int v = threadIdx.x; v < VV; v += blockDim.x) orow[v] = expf(orow[v] - M) * inv;
}

// ---------------------------------------------------------------------------
extern "C" void kernel_launch(void* const* d_in, const int* in_sizes, int n_in,
                              void* d_out, int out_size, void* d_ws, size_t ws_size,
                              hipStream_t stream) {
    const float* hidden    = (const float*)d_in[0];
    const float* embedding = (const float*)d_in[1];
    const float* w_ih      = (const float*)d_in[2];
    const float* w_hh      = (const float*)d_in[3];
    const float* b_ih      = (const float*)d_in[4];
    const float* b_hh      = (const float*)d_in[5];
    const float* w_out     = (const float*)d_in[6];
    const float* b_out     = (const float*)d_in[7];
    float* out = (float*)d_out;

    // workspace layout (floats)
    float* ws    = (float*)d_ws;
    float* hA    = ws;                        // 48*512
    float* hB    = hA    + RPAD * HH;         // 48*512
    float* embb  = hB    + RPAD * HH;         // 48*256
    float* rzbuf = embb  + RPAD * EE;         // 48*1024
    float* inbuf = rzbuf + RPAD * 2 * HH;     // 48*512
    float* hnbuf = inbuf + RPAD * HH;         // 48*512
    float* hhist = hnbuf + RPAD * HH;         // L*48*512
    float* pred  = hhist + (size_t)LL * RPAD * HH;  // 48*16000
    int* ibase    = (int*)(pred + (size_t)RPAD * VV);
    int* tok      = ibase;            // 40
    int* donehist = tok + ROWS;       // L*40
    int* bp       = donehist + LL * ROWS;  // L*40
    int* chosen   = bp + LL * ROWS;   // L*B

    init_kernel<<<RPAD, 256, 0, stream>>>(hidden, hA, tok);

    for (int i = 0; i < LL; ++i) {
        float* hh_i   = hhist + (size_t)i * RPAD * HH;
        int*   done_i = donehist + i * ROWS;
        prep_kernel<<<ROWS, 64, 0, stream>>>(embedding, tok, embb, done_i);
        gru_gemm_wmma<<<dim3(96, 3), 32, 0, stream>>>(embb, hA, w_ih, w_hh,
                                                      rzbuf, inbuf, hnbuf);
        gates_kernel<<<ROWS, 256, 0, stream>>>(rzbuf, inbuf, hnbuf, hA, b_ih, b_hh,
                                               hh_i, hB, done_i);
        logits_gemm_wmma<<<VV / 16, 32, 0, stream>>>(hh_i, w_out, b_out, pred);
        softmax_pred<<<ROWS, 256, 0, stream>>>(pred, done_i);
        topk_kernel<<<BB, 256, 0, stream>>>(pred, hA, hB, bp + i * ROWS, tok);
    }

    final_select<<<BB, 256, 0, stream>>>(pred, bp, chosen);
    reconstruct<<<LL * BB, 256, 0, stream>>>(hhist, chosen, donehist, w_out, b_out, out);
}